// G2SinkhornGate_31293131719431
// MI455X (gfx1250) — compile-verified
//
#include <hip/hip_runtime.h>
#include <stdint.h>

// Sinkhorn gate for MI455X (gfx1250, wave32).
// Factorized Sinkhorn: W = diag(rs) * A0 * diag(cs); A0 stays in registers.
// Single-wave workgroups (no barriers): column reduction is DPP8/DPP16
// butterflies (xor 1/2/4/8) + one v_wmma_f32_16x16x4_f32 per k that does the
// lane^16 reduction AND broadcasts the wave total to every lane.
// Persistent blocks async-prefetch the next 16KB tile into LDS (SADDR +
// immediate-offset form: zero per-issue address ALU / VGPRs) while the 20
// dependent iterations run from registers.

#define D_MODEL     512
#define KDIM        8
#define NITERS      20
#define EPSF        1e-8f
#define NTILES      (8 * 2048)
#define THREADS     32
#define RPT         (D_MODEL / THREADS)          // 16 rows per thread
#define TILE_FLOATS (D_MODEL * KDIM)             // 4096 floats = 16 KB
#define NBLOCKS     4096                         // 4 tiles per block

typedef __attribute__((ext_vector_type(2))) float v2f;
typedef __attribute__((ext_vector_type(8))) float v8f;

#define DPP8_SEL(a,b,c,d,e,f,g,h) \
  ((a)|((b)<<3)|((c)<<6)|((d)<<9)|((e)<<12)|((f)<<15)|((g)<<18)|((h)<<21))
#define SEL_XOR1 DPP8_SEL(1,0,3,2,5,4,7,6)
#define SEL_XOR2 DPP8_SEL(2,3,0,1,6,7,4,5)
#define SEL_XOR4 DPP8_SEL(4,5,6,7,0,1,2,3)

__device__ __forceinline__ float fast_rcp(float x) {
  return __builtin_amdgcn_rcpf(x);
}

template <unsigned SEL>
__device__ __forceinline__ float add_dpp8(float v) {
  int y = __builtin_amdgcn_mov_dpp8(__float_as_int(v), SEL);
  return v + __int_as_float(y);
}

// lane^8 within each 16-lane row: DPP16 row_ror:8 (0x128).
__device__ __forceinline__ float add_xor8(float v) {
  int y = __builtin_amdgcn_update_dpp(0, __float_as_int(v), 0x128, 0xf, 0xf, true);
  return v + __int_as_float(y);
}

// One b128 async global->LDS transfer; INST_OFFSET advances BOTH the LDS and
// the global address (ISA 8.2), so the whole tile copy shares one SGPR base,
// one global-offset VGPR and one LDS-address VGPR.
#define ASYNC_LD(OFF)                                                       \
  asm volatile("global_load_async_to_lds_b128 %0, %1, %2 offset:" #OFF      \
               :: "v"(lds_v), "v"(goff_v), "s"(gbase) : "memory")

// Async-copy one 16KB tile (4096 f32): 32 lanes x 32 b128, fully coalesced.
__device__ __forceinline__ void async_tile_load(uint64_t gbase,
                                                uint32_t lds_v,
                                                uint32_t goff_v) {
  ASYNC_LD(0);     ASYNC_LD(512);   ASYNC_LD(1024);  ASYNC_LD(1536);
  ASYNC_LD(2048);  ASYNC_LD(2560);  ASYNC_LD(3072);  ASYNC_LD(3584);
  ASYNC_LD(4096);  ASYNC_LD(4608);  ASYNC_LD(5120);  ASYNC_LD(5632);
  ASYNC_LD(6144);  ASYNC_LD(6656);  ASYNC_LD(7168);  ASYNC_LD(7680);
  ASYNC_LD(8192);  ASYNC_LD(8704);  ASYNC_LD(9216);  ASYNC_LD(9728);
  ASYNC_LD(10240); ASYNC_LD(10752); ASYNC_LD(11264); ASYNC_LD(11776);
  ASYNC_LD(12288); ASYNC_LD(12800); ASYNC_LD(13312); ASYNC_LD(13824);
  ASYNC_LD(14336); ASYNC_LD(14848); ASYNC_LD(15360); ASYNC_LD(15872);
}

__device__ __forceinline__ void wait_async_all() {
  asm volatile("s_wait_asynccnt 0" ::: "memory");
}

__device__ __forceinline__ void wait_ds_all() {
  asm volatile("s_wait_dscnt 0" ::: "memory");
}

__global__ __launch_bounds__(THREADS) void sinkhorn_gate_kernel(
    const float* __restrict__ in, float* __restrict__ out) {
  __shared__ __align__(16) float tile[TILE_FLOATS];  // 16 KB staging buffer

  const int tid = (int)threadIdx.x;
  const uint32_t lds_v  = (uint32_t)(uintptr_t)(void*)&tile[0] + (uint32_t)tid * 16u;
  const uint32_t goff_v = (uint32_t)tid * 16u;

  // WMMA constants: B = all ones (layout-independent), C = 0.
  v2f ones2; ones2[0] = 1.0f; ones2[1] = 1.0f;
  v8f czero = {0.f, 0.f, 0.f, 0.f, 0.f, 0.f, 0.f, 0.f};

  int tile_id = (int)blockIdx.x;
  if (tile_id < NTILES)
    async_tile_load((uint64_t)(uintptr_t)(in + (size_t)tile_id * TILE_FLOATS),
                    lds_v, goff_v);

  for (; tile_id < NTILES; tile_id += (int)gridDim.x) {
    // Wait for the prefetched tile (single wave: no barrier needed).
    wait_async_all();

    // LDS -> registers: this thread owns rows [tid*16, tid*16+16), 8 k each.
    float a[RPT][KDIM];
    {
      const float4* lt = (const float4*)tile;
#pragma unroll
      for (int j = 0; j < RPT; ++j) {
        float4 lo = lt[(tid * RPT + j) * 2 + 0];
        float4 hi = lt[(tid * RPT + j) * 2 + 1];
        a[j][0] = lo.x; a[j][1] = lo.y; a[j][2] = lo.z; a[j][3] = lo.w;
        a[j][4] = hi.x; a[j][5] = hi.y; a[j][6] = hi.z; a[j][7] = hi.w;
      }
    }
    // All DS reads must land before the async engine may overwrite the
    // buffer (ASYNC and DS are mutually unordered).
    wait_ds_all();

    // Prefetch the next tile; it streams from HBM while we run 20
    // dependent Sinkhorn iterations out of registers.
    {
      int next = tile_id + (int)gridDim.x;
      if (next < NTILES)
        async_tile_load((uint64_t)(uintptr_t)(in + (size_t)next * TILE_FLOATS),
                        lds_v, goff_v);
    }

    float cs[KDIM];                 // column scales (replicated per lane)
    float rs[RPT];                  // row scales for this thread's rows
#pragma unroll
    for (int k = 0; k < KDIM; ++k) cs[k] = 1.0f;
#pragma unroll
    for (int j = 0; j < RPT; ++j)  rs[j] = 1.0f;

    for (int it = 0; it < NITERS; ++it) {
      // ---- column step: T[k] = sum_r a[r][k] * rs[r] over all 512 rows ----
      float t[KDIM];
#pragma unroll
      for (int k = 0; k < KDIM; ++k) {
        float acc = 0.0f;
#pragma unroll
        for (int j = 0; j < RPT; ++j) acc = fmaf(a[j][k], rs[j], acc);
        t[k] = acc;
      }
      // Butterflies xor 1/2/4 via DPP8, xor 8 via DPP16 row_ror:8 —
      // pure VALU, no LDS traffic. Each 16-lane half is fully reduced.
#pragma unroll
      for (int k = 0; k < KDIM; ++k) {
        float v = t[k];
        v = add_dpp8<SEL_XOR1>(v);
        v = add_dpp8<SEL_XOR2>(v);
        v = add_dpp8<SEL_XOR4>(v);
        v = add_xor8(v);
        t[k] = v;
      }
      // WMMA lane^16 reduce + broadcast, consumed immediately:
      // D[m][n] = t@lane_m + t@lane_(m+16) = wave total in every lane.
#pragma unroll
      for (int k = 0; k < KDIM; ++k) {
        v2f aa; aa[0] = t[k]; aa[1] = 0.0f;
        v8f d = __builtin_amdgcn_wmma_f32_16x16x4_f32(
            false, aa, false, ones2, (short)0, czero, false, false);
        float T = d[0];                              // = sum_r a*rs
        // colsum = cs*T ; cs' = cs / (colsum + eps)
        cs[k] = cs[k] * fast_rcp(fmaf(T, cs[k], EPSF));
      }
      // ---- row step: rowsum = rs * sum_k a*cs' ; rs' = rs/(rowsum+eps) ----
#pragma unroll
      for (int j = 0; j < RPT; ++j) {
        float m = 0.0f;
#pragma unroll
        for (int k = 0; k < KDIM; ++k) m = fmaf(a[j][k], cs[k], m);
        rs[j] = rs[j] * fast_rcp(fmaf(m, rs[j], EPSF));
      }
    }

    // out[r] = sum_k W*A0 = rs[r] * sum_k a^2 * cs[k]
    float4* op = (float4*)(out + (size_t)tile_id * D_MODEL + tid * RPT);
#pragma unroll
    for (int q = 0; q < RPT / 4; ++q) {
      float o[4];
#pragma unroll
      for (int u = 0; u < 4; ++u) {
        int j = q * 4 + u;
        float acc = 0.0f;
#pragma unroll
        for (int k = 0; k < KDIM; ++k) acc = fmaf(a[j][k] * a[j][k], cs[k], acc);
        o[u] = acc * rs[j];
      }
      op[q] = make_float4(o[0], o[1], o[2], o[3]);
    }
  }
}

extern "C" void kernel_launch(void* const* d_in, const int* in_sizes, int n_in,
                              void* d_out, int out_size, void* d_ws, size_t ws_size,
                              hipStream_t stream) {
  (void)in_sizes; (void)n_in; (void)out_size; (void)d_ws; (void)ws_size;
  const float* in = (const float*)d_in[0];
  float* out = (float*)d_out;
  sinkhorn_gate_kernel<<<dim3(NBLOCKS), dim3(THREADS), 0, stream>>>(in, out);
}